// CustomLSTM_8461085573201
// MI455X (gfx1250) — compile-verified
//
#include <hip/hip_runtime.h>
#include <math.h>

// tanh(X[:, T-1, :] @ W + bias)   X:(256,512,256) f32, W:(256,256) f32, bias:(256) f32
// Full-fp32 matrix path via V_WMMA_F32_16X16X4_F32 (CDNA5, wave32).

typedef __attribute__((ext_vector_type(2))) float v2f;
typedef __attribute__((ext_vector_type(4))) float v4f;
typedef __attribute__((ext_vector_type(8))) float v8f;

#define B_DIM 256
#define T_DIM 512
#define F_DIM 256
#define U_DIM 256

__global__ __launch_bounds__(256)
void lstm_head_tanh_wmma_f32(const float* __restrict__ x,
                             const float* __restrict__ w,
                             const float* __restrict__ bias,
                             float* __restrict__ out)
{
    // One block = one 16-wide N stripe x one half of M (8 tiles of 16 rows).
    // 8 waves per block, one 16x16 output tile per wave.
    __shared__ __align__(16) float Bs[F_DIM * 16];   // W[:, n0:n0+16], 16 KB

    const int tid  = threadIdx.x;
    const int wave = tid >> 5;          // 0..7
    const int lane = tid & 31;          // wave32
    const int half = lane >> 4;         // 0: lanes 0-15, 1: lanes 16-31
    const int lr   = lane & 15;

    const int n0 = blockIdx.x * 16;     // N tile origin

    // ---- Stage B column tile into LDS (coalesced float4, 4 iters/thread) ----
    {
        const v4f* __restrict__ wrow;
        for (int i = tid; i < F_DIM * 4; i += 256) {
            const int row = i >> 2;     // K row 0..255
            const int q   = i & 3;      // float4 within the 16-wide stripe
            wrow = reinterpret_cast<const v4f*>(w + row * U_DIM + n0); // 64B aligned
            reinterpret_cast<v4f*>(Bs)[i] = wrow[q];
        }
    }
    __syncthreads();

    // ---- Per-wave tile: rows [mt*16, mt*16+16) of the output ----
    const int    mt        = blockIdx.y * 8 + wave;                  // 0..15
    const size_t rowStride = (size_t)T_DIM * F_DIM;                  // stride between batch rows
    const float* __restrict__ rowA =
        x + (size_t)(mt * 16 + lr) * rowStride + (size_t)(T_DIM - 1) * F_DIM;

    // Prefetch this lane's 1KB X row (t = T-1 slice) into the cache hierarchy.
    #pragma unroll
    for (int pb = 0; pb < 1024; pb += 256)
        __builtin_prefetch(reinterpret_cast<const char*>(rowA) + pb, 0, 0);

    v8f acc = {};

    // A 16x4 f32 frag: v0 = K(2*half), v1 = K(2*half+1)  -> contiguous float2 per lane.
    // B 4x16 f32 frag: lane = N, v0/v1 = K(2*half), K(2*half+1) from LDS.
    #pragma unroll 4
    for (int kk = 0; kk < F_DIM; kk += 4) {
        const int kb = kk + half * 2;
        v2f a = *reinterpret_cast<const v2f*>(rowA + kb);   // 8B-aligned
        v2f b;
        b.x = Bs[kb * 16 + lr];
        b.y = Bs[(kb + 1) * 16 + lr];
        acc = __builtin_amdgcn_wmma_f32_16x16x4_f32(
            /*neg_a=*/false, a, /*neg_b=*/false, b,
            /*c_mod=*/(short)0, acc, /*reuse_a=*/false, /*reuse_b=*/false);
    }

    // ---- Epilogue: bias + tanh straight from the C/D VGPR layout ----
    // VGPR r: lanes 0-15 -> M = r, lanes 16-31 -> M = r+8; lane -> N = lr.
    const int   col = n0 + lr;
    const float bv  = bias[col];
    #pragma unroll
    for (int r = 0; r < 8; ++r) {
        const int row = mt * 16 + r + half * 8;
        out[row * U_DIM + col] = tanhf(acc[r] + bv);
    }
}

extern "C" void kernel_launch(void* const* d_in, const int* in_sizes, int n_in,
                              void* d_out, int out_size, void* d_ws, size_t ws_size,
                              hipStream_t stream) {
    (void)in_sizes; (void)n_in; (void)out_size; (void)d_ws; (void)ws_size;
    const float* x    = (const float*)d_in[0];   // (256, 512, 256)
    const float* w    = (const float*)d_in[1];   // (256, 256)
    const float* bias = (const float*)d_in[2];   // (256,)
    float*       out  = (float*)d_out;           // (256, 256)

    dim3 grid(U_DIM / 16, 2);   // 16 N-stripes x 2 M-halves = 32 blocks, 256 waves
    dim3 block(256);            // 8 waves (wave32)
    lstm_head_tanh_wmma_f32<<<grid, block, 0, stream>>>(x, w, bias, out);
}